// Phngb_38474317037901
// MI455X (gfx1250) — compile-verified
//
#include <hip/hip_runtime.h>

typedef __attribute__((ext_vector_type(2))) float v2f;
typedef __attribute__((ext_vector_type(8))) float v8f;

#define NFEAT 4096
#define DCOORD 64
#define NBATCH 4096
#define KNB 8
#define NCOLS (NFEAT * KNB + 1)   // 32769

// ---------------------------------------------------------------------------
// Stage 0: per-feature squared norm  sq[f] = sum_d coord[d][f]^2
// ---------------------------------------------------------------------------
__global__ void sq_kernel(const float* __restrict__ coord, float* __restrict__ sq) {
    int f = blockIdx.x * blockDim.x + threadIdx.x;
    if (f < NFEAT) {
        float s = 0.0f;
        for (int d = 0; d < DCOORD; ++d) {
            float x = coord[d * NFEAT + f];
            s = fmaf(x, x, s);
        }
        sq[f] = s;
    }
}

// ---------------------------------------------------------------------------
// Stage 1: WMMA Gram tiles + stable per-row top-8 of clamped squared distance
// One workgroup (4 waves) per 16-feature row block. B tile is preloaded into
// a register array so the 32 loads issue as one clause (single wait) and the
// 16 chained v_wmma_f32_16x16x4_f32 run back-to-back.
// ---------------------------------------------------------------------------
__global__ __launch_bounds__(128) void knn_kernel(const float* __restrict__ coord,
                                                  const float* __restrict__ sq,
                                                  int* __restrict__ nbr) {
    __shared__ float stage[4 * 256];     // per-wave 16x16 dsq tile
    __shared__ float lv[4][16][KNB];     // per-wave per-row sorted top-8 values
    __shared__ int   li[4][16][KNB];     // matching indices

    const int tid   = threadIdx.x;
    const int w     = tid >> 5;          // wave id (wave32)
    const int lane  = tid & 31;
    const int half  = lane >> 4;         // 0: lanes 0-15, 1: lanes 16-31
    const int nl    = lane & 15;
    const int ibase = blockIdx.x * 16;

    // init per-wave lists
    if (lane < 16) {
        for (int s = 0; s < KNB; ++s) {
            lv[w][lane][s] = 3.4e38f;
            li[w][lane][s] = 0x7fffffff;
        }
    }

    // Preload A operand once: 16x64 f32 row block, 16 k-steps of 16x4.
    // 32-bit A 16x4 layout: lanes 0-15 hold M=0..15 {K, K+1}; lanes 16-31 hold {K+2, K+3}.
    v2f A[16];
#pragma unroll
    for (int ks = 0; ks < 16; ++ks) {
        int kb = ks * 4 + half * 2;
        A[ks].x = coord[(kb + 0) * NFEAT + ibase + nl];
        A[ks].y = coord[(kb + 1) * NFEAT + ibase + nl];
    }

    // Per-lane row norms for this lane's C-tile rows (M = v + 8*half)
    float sqi[8];
#pragma unroll
    for (int v = 0; v < 8; ++v) sqi[v] = sq[ibase + v + half * 8];

    for (int iter = 0; iter < 64; ++iter) {
        const int jbase = (iter * 4 + w) * 16;
        const float sqj = sq[jbase + nl];

        // Load the whole B tile (4x16 per k-step, 16 steps) into registers
        // first: 32 coalesced b32 loads in one clause, then one wait.
        v2f Bv[16];
#pragma unroll
        for (int ks = 0; ks < 16; ++ks) {
            int kb = ks * 4 + half * 2;
            Bv[ks].x = coord[(kb + 0) * NFEAT + jbase + nl];
            Bv[ks].y = coord[(kb + 1) * NFEAT + jbase + nl];
        }

        v8f acc = {};
#pragma unroll
        for (int ks = 0; ks < 16; ++ks) {
            acc = __builtin_amdgcn_wmma_f32_16x16x4_f32(
                /*neg_a=*/false, A[ks], /*neg_b=*/false, Bv[ks],
                /*c_mod=*/(short)0, acc, /*reuse_a=*/false, /*reuse_b=*/false);
        }

        // C/D layout: VGPR v, lanes 0-15 -> M=v, N=lane; lanes 16-31 -> M=v+8, N=lane-16
#pragma unroll
        for (int v = 0; v < 8; ++v) {
            int M = v + half * 8;
            float dsq = fmaxf(sqi[v] + sqj - 2.0f * acc[v], 0.0f);
            stage[w * 256 + M * 16 + nl] = dsq;
        }
        __syncthreads();

        // stable insertion: lane r of each wave owns row r; scan n ascending (j monotone)
        if (lane < 16) {
            const int r = lane;
            float* LV = &lv[w][r][0];
            int*   LI = &li[w][r][0];
            for (int n = 0; n < 16; ++n) {
                float v = stage[w * 256 + r * 16 + n];
                int   j = jbase + n;
                if (v < LV[KNB - 1] || (v == LV[KNB - 1] && j < LI[KNB - 1])) {
                    int p = KNB - 1;
                    while (p > 0 && (LV[p - 1] > v || (LV[p - 1] == v && LI[p - 1] > j))) {
                        LV[p] = LV[p - 1];
                        LI[p] = LI[p - 1];
                        --p;
                    }
                    LV[p] = v;
                    LI[p] = j;
                }
            }
        }
        __syncthreads();
    }

    // Merge 4 sorted lists per row (stable lexicographic (value, index) selection)
    if (tid < 16) {
        const int r = tid;
        int p[4] = {0, 0, 0, 0};
        for (int s = 0; s < KNB; ++s) {
            float bestv = 3.5e38f;
            int   besti = 0x7fffffff;
            int   bw    = 0;
            for (int ww = 0; ww < 4; ++ww) {
                if (p[ww] < KNB) {
                    float vv = lv[ww][r][p[ww]];
                    int   ii = li[ww][r][p[ww]];
                    if (vv < bestv || (vv == bestv && ii < besti)) {
                        bestv = vv; besti = ii; bw = ww;
                    }
                }
            }
            nbr[(ibase + r) * KNB + s] = besti;
            p[bw]++;
        }
    }
}

// ---------------------------------------------------------------------------
// Stage 2: bandwidth-dominant gather. Consecutive lanes -> consecutive output
// columns (coalesced b32 stores). Input row (16 KB) and nbr (128 KB) stay hot
// in cache while a row's column-blocks execute.
// ---------------------------------------------------------------------------
__global__ void gather_kernel(const float* __restrict__ inputs,
                              const int* __restrict__ nbr,
                              float* __restrict__ out) {
    const int c = blockIdx.x * blockDim.x + threadIdx.x;
    const int b = blockIdx.y;
    if (c < NCOLS) {
        const int j = (c == 0) ? 0 : nbr[c - 1];
        out[(size_t)b * NCOLS + c] = inputs[b * NFEAT + j];
    }
}

extern "C" void kernel_launch(void* const* d_in, const int* in_sizes, int n_in,
                              void* d_out, int out_size, void* d_ws, size_t ws_size,
                              hipStream_t stream) {
    const float* inputs = (const float*)d_in[0];   // [4096, 4096]
    const float* coord  = (const float*)d_in[1];   // [64, 4096]
    float* out = (float*)d_out;                    // [4096, 32769]

    float* sq  = (float*)d_ws;                                   // 4096 floats
    int*   nbr = (int*)((char*)d_ws + NFEAT * sizeof(float));    // 4096*8 ints

    sq_kernel<<<dim3(NFEAT / 256), 256, 0, stream>>>(coord, sq);
    knn_kernel<<<dim3(NFEAT / 16), 128, 0, stream>>>(coord, sq, nbr);
    gather_kernel<<<dim3((NCOLS + 255) / 256, NBATCH), 256, 0, stream>>>(inputs, nbr, out);
}